// LightGCN_71270687310420
// MI455X (gfx1250) — compile-verified
//
#include <hip/hip_runtime.h>
#include <hip/hip_bf16.h>
#include <stdint.h>

#define LGCN_DIM 64
#define WAVES_PER_BLOCK 8
#define TILE_EDGES 32

typedef __attribute__((ext_vector_type(2))) float v2f;
typedef __attribute__((ext_vector_type(8))) float v8f;

// ---------------------------------------------------------------------------
// CDNA5 async global->LDS helpers (inline asm; portable across toolchains).
// Generic pointers to LDS carry the workgroup-relative LDS byte offset in
// their low 32 bits (aperture addressing), which is exactly what the VDST
// operand of global_load_async_to_lds_* expects.
// ---------------------------------------------------------------------------
__device__ __forceinline__ uint32_t lds_off32(const void* p) {
  return (uint32_t)(uintptr_t)p;
}

__device__ __forceinline__ void async_ld_dword(uint32_t lds_off, const void* gaddr) {
  asm volatile("global_load_async_to_lds_b32 %0, %1, off"
               :: "v"(lds_off), "v"(gaddr)
               : "memory");
}

__device__ __forceinline__ void wait_async_le3() {
  asm volatile("s_wait_asynccnt 0x3" ::: "memory");
}
__device__ __forceinline__ void wait_async_0() {
  asm volatile("s_wait_asynccnt 0x0" ::: "memory");
}

// ---------------------------------------------------------------------------
// x0 build: xA = concat(user_emb, item_emb); out = scale * x0; xB = 0
// ---------------------------------------------------------------------------
__global__ void __launch_bounds__(256)
lgcn_init(const float* __restrict__ ue, const float* __restrict__ ie,
          float* __restrict__ xA, float* __restrict__ xB,
          float* __restrict__ out, float scale,
          long long uElems, long long nElems) {
  const long long n4 = nElems >> 2;
  const long long u4 = uElems >> 2;
  for (long long i = (long long)blockIdx.x * blockDim.x + threadIdx.x; i < n4;
       i += (long long)gridDim.x * blockDim.x) {
    float4 v = (i < u4) ? ((const float4*)ue)[i] : ((const float4*)ie)[i - u4];
    ((float4*)xA)[i] = v;
    ((float4*)out)[i] = make_float4(v.x * scale, v.y * scale,
                                    v.z * scale, v.w * scale);
    ((float4*)xB)[i] = make_float4(0.f, 0.f, 0.f, 0.f);
  }
}

// ---------------------------------------------------------------------------
// out += scale * x, expressed per 16x16 f32 tile as chained WMMA:
//   D = A_j(16x4 slice of x-tile) x B_j(4x16 slice of scale*I16) + C
// for j = 0..3 (K walks 0..15), C seeded from the out-tile.
// Also (optionally) zeroes the buffer targeted by the next SpMM layer.
// VGPR layouts per CDNA5 ISA 7.12.2:
//   A 16x4 : lanes 0-15 -> (M=lane, K=0/1), lanes 16-31 -> K=2/3
//   B 4x16 : VGPR0 = rows K=0 / K=2, VGPR1 = rows K=1 / K=3 (lane halves)
//   C/D    : VGPR i = rows M=i (lanes 0-15) / M=i+8 (lanes 16-31)
// ---------------------------------------------------------------------------
__global__ void __launch_bounds__(256)
lgcn_acc_wmma(float* __restrict__ out, const float* __restrict__ x,
              float* __restrict__ zbuf, float scale, long long nElems) {
  const int lane = threadIdx.x & 31;
  const int wv   = threadIdx.x >> 5;
  const long long waveId = (long long)blockIdx.x * WAVES_PER_BLOCK + wv;
  const long long nWaves = (long long)gridDim.x * WAVES_PER_BLOCK;
  const long long nTiles = nElems >> 8;   // 256 floats per 16x16 tile

  const int half = lane >> 4;   // 0: lanes 0-15, 1: lanes 16-31
  const int n    = lane & 15;   // column (and M for A loads)

  for (long long t = waveId; t < nTiles; t += nWaves) {
    const float* xt = x + (t << 8);
    float* ot       = out + (t << 8);

    v8f c;
#pragma unroll
    for (int i = 0; i < 8; ++i)
      c[i] = ot[(i + half * 8) * 16 + n];

#pragma unroll
    for (int j = 0; j < 4; ++j) {
      v2f a;
      a.x = xt[n * 16 + 4 * j + 2 * half + 0];
      a.y = xt[n * 16 + 4 * j + 2 * half + 1];
      v2f b;
      b.x = (n == 4 * j + 2 * half + 0) ? scale : 0.f;
      b.y = (n == 4 * j + 2 * half + 1) ? scale : 0.f;
      c = __builtin_amdgcn_wmma_f32_16x16x4_f32(false, a, false, b,
                                                (short)0, c, false, false);
    }

#pragma unroll
    for (int i = 0; i < 8; ++i)
      ot[(i + half * 8) * 16 + n] = c[i];

    if (zbuf) {
      float4* z4 = (float4*)(zbuf + (t << 8));
      z4[lane]      = make_float4(0.f, 0.f, 0.f, 0.f);
      z4[lane + 32] = make_float4(0.f, 0.f, 0.f, 0.f);
    }
  }

  // scalar tail (nElems not a multiple of 256)
  const long long tail0 = nTiles << 8;
  const long long gtid  = (long long)blockIdx.x * blockDim.x + threadIdx.x;
  const long long tstr  = (long long)gridDim.x * blockDim.x;
  for (long long i = tail0 + gtid; i < nElems; i += tstr) {
    out[i] += scale * x[i];
    if (zbuf) zbuf[i] = 0.f;
  }
}

// ---------------------------------------------------------------------------
// SpMM: xcur[row[e]] += vals[e] * xprev[col[e]]   (64 floats per row)
// One wave32 per edge; float2 per lane (256B coalesced row access).
// Edge metadata (row/col/val) is streamed via async global->LDS loads,
// double-buffered per wave (3 async dwords per 32-edge tile).
// ---------------------------------------------------------------------------
__global__ void __launch_bounds__(256)
lgcn_spmm(const float* __restrict__ xprev, float* __restrict__ xcur,
          const int* __restrict__ row, const int* __restrict__ col,
          const float* __restrict__ vals, long long nnz) {
  __shared__ int   sRow[WAVES_PER_BLOCK][2][TILE_EDGES];
  __shared__ int   sCol[WAVES_PER_BLOCK][2][TILE_EDGES];
  __shared__ float sVal[WAVES_PER_BLOCK][2][TILE_EDGES];

  const int lane = threadIdx.x & 31;
  const int wv   = threadIdx.x >> 5;
  const long long waveId = (long long)blockIdx.x * WAVES_PER_BLOCK + wv;
  const long long nWaves = (long long)gridDim.x * WAVES_PER_BLOCK;
  const long long nTiles = (nnz + TILE_EDGES - 1) / TILE_EDGES;

  auto issue = [&](long long t, int b) {
    long long e = t * TILE_EDGES + lane;
    if (e > nnz - 1) e = nnz - 1;           // clamp so tail tiles stay in-bounds
    async_ld_dword(lds_off32(&sRow[wv][b][lane]), row  + e);
    async_ld_dword(lds_off32(&sCol[wv][b][lane]), col  + e);
    async_ld_dword(lds_off32(&sVal[wv][b][lane]), vals + e);
  };

  int buf = 0;
  const long long t0 = waveId;
  if (t0 < nTiles) issue(t0, 0);

  for (long long t = t0; t < nTiles; t += nWaves) {
    const long long tn = t + nWaves;
    const bool havenext = (tn < nTiles);
    if (havenext) {
      issue(tn, buf ^ 1);   // prefetch next tile's metadata (3 outstanding)
      wait_async_le3();     // current tile's 3 loads done (in-order retire)
    } else {
      wait_async_0();
    }

    // Per-lane copy of this tile's metadata (3 LDS loads per lane total).
    const int   myR = sRow[wv][buf][lane];
    const int   myC = sCol[wv][buf][lane];
    const float myV = sVal[wv][buf][lane];

    // Warm L2/WGP$ for all 32 gathered rows of this tile (256B = 2x128B).
    {
      const char* p = (const char*)(xprev + ((size_t)myC << 6));
      __builtin_prefetch(p, 0, 1);
      __builtin_prefetch(p + 128, 0, 1);
    }

    const long long base = t * TILE_EDGES;
    const int kmax = (int)((nnz - base < TILE_EDGES) ? (nnz - base) : TILE_EDGES);
    for (int k = 0; k < kmax; ++k) {
      const int   r = __shfl(myR, k, 32);
      const int   c = __shfl(myC, k, 32);
      const float v = __shfl(myV, k, 32);
      const float2 xv = ((const float2*)(xprev + ((size_t)c << 6)))[lane];
      float* dst = xcur + ((size_t)r << 6) + (lane << 1);
      unsafeAtomicAdd(dst + 0, v * xv.x);   // global_atomic_add_f32, no return
      unsafeAtomicAdd(dst + 1, v * xv.y);
    }
    buf ^= 1;
  }
}

// ---------------------------------------------------------------------------
// Host side: 1 init + 3 x (spmm + wmma-acc). All launches on `stream`.
// out accumulates 0.25 * x_l per layer (so no final rescale pass).
// ---------------------------------------------------------------------------
extern "C" void kernel_launch(void* const* d_in, const int* in_sizes, int n_in,
                              void* d_out, int out_size, void* d_ws, size_t ws_size,
                              hipStream_t stream) {
  const float* ue   = (const float*)d_in[0];
  const float* ie   = (const float*)d_in[1];
  const int*   row  = (const int*)d_in[2];
  const int*   col  = (const int*)d_in[3];
  const float* vals = (const float*)d_in[4];

  const long long uElems = in_sizes[0];            // U * 64
  const long long iElems = in_sizes[1];            // I * 64
  const long long nnz    = in_sizes[2];
  const long long nElems = uElems + iElems;        // N * 64
  const float s = 1.0f / 4.0f;                     // 1 / (N_LAYERS + 1)

  float* A   = (float*)d_ws;                       // ping
  float* B   = A + nElems;                         // pong
  float* out = (float*)d_out;

  const int tb = 256;
  const long long n4 = nElems >> 2;
  int gInit = (int)((n4 + tb - 1) / tb);
  if (gInit < 1) gInit = 1;

  const long long accTiles = nElems >> 8;
  long long gAcc64 = (accTiles + WAVES_PER_BLOCK - 1) / WAVES_PER_BLOCK;
  if (gAcc64 > 16384) gAcc64 = 16384;
  if (gAcc64 < 1) gAcc64 = 1;
  const int gAcc = (int)gAcc64;

  const long long spTiles = (nnz + TILE_EDGES - 1) / TILE_EDGES;
  long long gSp64 = (spTiles + WAVES_PER_BLOCK - 1) / WAVES_PER_BLOCK;
  if (gSp64 > 2048) gSp64 = 2048;
  if (gSp64 < 1) gSp64 = 1;
  const int gSp = (int)gSp64;

  // out = s*x0 ; A = x0 ; B = 0
  lgcn_init<<<gInit, tb, 0, stream>>>(ue, ie, A, B, out, s, uElems, nElems);

  // layer 1: B = S*A ; out += s*B ; A = 0
  lgcn_spmm    <<<gSp, tb, 0, stream>>>(A, B, row, col, vals, nnz);
  lgcn_acc_wmma<<<gAcc, tb, 0, stream>>>(out, B, A, s, nElems);

  // layer 2: A = S*B ; out += s*A ; B = 0
  lgcn_spmm    <<<gSp, tb, 0, stream>>>(B, A, row, col, vals, nnz);
  lgcn_acc_wmma<<<gAcc, tb, 0, stream>>>(out, A, B, s, nElems);

  // layer 3: B = S*A ; out += s*B
  lgcn_spmm    <<<gSp, tb, 0, stream>>>(A, B, row, col, vals, nnz);
  lgcn_acc_wmma<<<gAcc, tb, 0, stream>>>(out, B, nullptr, s, nElems);
}